// APoTQuantizer_23948737642755
// MI455X (gfx1250) — compile-verified
//
#include <hip/hip_runtime.h>
#include <math.h>

typedef __attribute__((ext_vector_type(4))) float v4f;

#define APOT_MAX_LEVELS 1024
#define BLOCK 256

__global__ __launch_bounds__(BLOCK) void apot_quant_kernel(
    const float* __restrict__ x,
    const float* __restrict__ alpha_p,
    const float* __restrict__ levels,
    float* __restrict__ out,
    int n_elem, int n_levels, float g)
{
    __shared__ float s_levels[APOT_MAX_LEVELS];

    const int tid = threadIdx.x;

    // ---- Stage the level table into LDS via CDNA5 async global->LDS DMA ----
    // Low 32 bits of a flat pointer into the shared aperture == LDS byte offset
    for (int i = tid; i < n_levels; i += BLOCK) {
        unsigned lds_off = (unsigned)(unsigned long long)(const void*)&s_levels[i];
        const float* gp = levels + i;
        asm volatile("global_load_async_to_lds_b32 %0, %1, off"
                     :: "v"(lds_off), "v"(gp)
                     : "memory");
    }
    asm volatile("s_wait_asynccnt 0" ::: "memory");
    __syncthreads();

    // ---- Uniform per-dispatch scalars (match reference fp32 rounding) ----
    const float alpha = alpha_p[0];
    const float t = alpha * g;             // exact: g = 2^-13
    float a = (alpha - t) + t;             // reference's grad-scale identity, fp32
    a = fmaxf(a, 1e-8f);
    const int nl = n_levels;

    // Per-element finish after lower_bound: exact reference semantics.
    auto finish = [&](float xn, int pos) -> float {
        int p0 = (pos - 1 < 0) ? 0 : pos - 1;     // clip(pos-1, 0, n-1)
        int p1 = (pos > nl - 1) ? nl - 1 : pos;   // clip(pos,   0, n-1)
        float l0 = s_levels[p0];
        float l1 = s_levels[p1];
        float q = (fabsf(xn - l0) > fabsf(xn - l1)) ? l1 : l0;  // ties -> l0
        return q * a;
    };

    // Quantize 4 elements with fused, branchless, uniform-trip lower_bound:
    // all 4 ds gathers per iteration are adjacent (clauseable, ILP on DS).
    auto quant4 = [&](v4f v) -> v4f {
        float x0 = fminf(fmaxf(v.x, -a), a) / a;   // clip + IEEE divide
        float x1 = fminf(fmaxf(v.y, -a), a) / a;
        float x2 = fminf(fmaxf(v.z, -a), a) / a;
        float x3 = fminf(fmaxf(v.w, -a), a) / a;
        int b0 = 0, b1 = 0, b2 = 0, b3 = 0;
        int len = nl;
        while (len > 1) {                 // trip count uniform in nl (8 for nl=155)
            int half = len >> 1;
            float t0 = s_levels[b0 + half - 1];
            float t1 = s_levels[b1 + half - 1];
            float t2 = s_levels[b2 + half - 1];
            float t3 = s_levels[b3 + half - 1];
            b0 = (t0 < x0) ? b0 + half : b0;
            b1 = (t1 < x1) ? b1 + half : b1;
            b2 = (t2 < x2) ? b2 + half : b2;
            b3 = (t3 < x3) ? b3 + half : b3;
            len -= half;
        }
        int p0 = b0 + ((s_levels[b0] < x0) ? 1 : 0);   // searchsorted 'left'
        int p1 = b1 + ((s_levels[b1] < x1) ? 1 : 0);
        int p2 = b2 + ((s_levels[b2] < x2) ? 1 : 0);
        int p3 = b3 + ((s_levels[b3] < x3) ? 1 : 0);
        v4f r;
        r.x = finish(x0, p0);
        r.y = finish(x1, p1);
        r.z = finish(x2, p2);
        r.w = finish(x3, p3);
        return r;
    };

    // Scalar path for the tail.
    auto quant1 = [&](float xv) -> float {
        float xn = fminf(fmaxf(xv, -a), a) / a;
        int b = 0, len = nl;
        while (len > 1) {
            int half = len >> 1;
            b = (s_levels[b + half - 1] < xn) ? b + half : b;
            len -= half;
        }
        int pos = b + ((s_levels[b] < xn) ? 1 : 0);
        return finish(xn, pos);
    };

    const int gtid   = blockIdx.x * BLOCK + tid;
    const int stride = gridDim.x * BLOCK;
    const int n4     = n_elem >> 2;

    const v4f* __restrict__ x4 = (const v4f*)x;
    v4f* __restrict__ o4       = (v4f*)out;

    // ---- Streaming main loop: B128 nontemporal load/store ----
    for (int idx = gtid; idx < n4; idx += stride) {
        v4f v = __builtin_nontemporal_load(&x4[idx]);
        v4f r = quant4(v);
        __builtin_nontemporal_store(r, &o4[idx]);
    }

    // ---- Scalar tail (n_elem not divisible by 4) ----
    for (int idx = (n4 << 2) + gtid; idx < n_elem; idx += stride) {
        out[idx] = quant1(x[idx]);
    }
}

extern "C" void kernel_launch(void* const* d_in, const int* in_sizes, int n_in,
                              void* d_out, int out_size, void* d_ws, size_t ws_size,
                              hipStream_t stream) {
    const float* x      = (const float*)d_in[0];
    const float* alpha  = (const float*)d_in[1];
    const float* levels = (const float*)d_in[2];
    float* out          = (float*)d_out;

    const int n_elem   = in_sizes[0];
    const int n_levels = in_sizes[2];

    // Reference: g = 1 / sqrt(numel); for 4096*16384 this is exactly 2^-13.
    const float g = (float)(1.0 / sqrt((double)n_elem));

    int n4 = n_elem >> 2;
    int blocks = (n4 + BLOCK - 1) / BLOCK;
    if (blocks > 8192) blocks = 8192;   // grid-stride; amortize LDS staging
    if (blocks < 1) blocks = 1;

    apot_quant_kernel<<<dim3(blocks), dim3(BLOCK), 0, stream>>>(
        x, alpha, levels, out, n_elem, n_levels, g);
}